// SSKernelSlow_28853590294574
// MI455X (gfx1250) — compile-verified
//
#include <hip/hip_runtime.h>

// SSKernelSlow on MI455X (gfx1250, wave32, WMMA):
//   dA = 2*(I - dt/2 A)^{-1} - I   (identity avoids a second solve)
//   k[h,l] = C . dA^l . dB, computed as: first 16 columns by serial matvec,
//   P = dA^16 by 4 WMMA squarings, then 63 block steps X <- P @ X with
//   V_WMMA_F32_16X16X4_F32 (f32 tensor op: needed, dA is powered to 1024).
// Scratch layout in d_ws: buf0 (64MB) | buf1 (64MB) | X0 (4MB)  => 132 MB.

typedef __attribute__((ext_vector_type(2))) float v2f;
typedef __attribute__((ext_vector_type(8))) float v8f;

#define H_    256
#define N_    256
#define L_    1024
#define MSTR  257   // padded LDS row stride for 256x256 matrices (bank-conflict free)
#define XSTR  17    // padded LDS row stride for 256x16 state block

// ---------------------------------------------------------------------------
// Kernel A: per-head discretization. One workgroup (8 waves) per head.
// LDS: 256x257 f32 matrix (263 KB) -- only possible with CDNA5's 320KB WGP LDS.
// ---------------------------------------------------------------------------
__global__ __launch_bounds__(256) void disc_invert_kernel(
    const float* __restrict__ A, const float* __restrict__ Bv,
    const float* __restrict__ Cv, const float* __restrict__ logdt,
    float* __restrict__ dA_out, float* __restrict__ X0_out,
    float* __restrict__ out)
{
    __shared__ float mat[N_ * MSTR];
    __shared__ float bvec[N_];
    __shared__ float cvec[N_];
    __shared__ float xv[N_];
    __shared__ float red[N_];

    const int h   = blockIdx.x;
    const int tid = threadIdx.x;
    const float dt = expf(logdt[h]);
    const float hd = 0.5f * dt;
    const float* __restrict__ Ah = A + (size_t)h * N_ * N_;

    // mat = I - dt/2 * A
    for (int idx = tid; idx < N_ * N_; idx += 256) {
        const int i = idx >> 8, j = idx & 255;
        mat[i * MSTR + j] = (i == j ? 1.0f : 0.0f) - hd * Ah[idx];
    }
    bvec[tid] = Bv[h * N_ + tid];
    cvec[tid] = Cv[h * N_ + tid];   // CH == 1
    __syncthreads();

    // In-place Gauss-Jordan inverse (no pivoting: matrix is ~ (1+dt/2) I,
    // strongly diagonally dominant).
    for (int k = 0; k < N_; ++k) {
        const int krow = k * MSTR;
        const float ip = 1.0f / mat[krow + k];
        if (tid != k) mat[krow + tid] *= ip;        // scale pivot row
        __syncthreads();
        if (tid != k) {                              // eliminate: thread owns row tid
            const int trow = tid * MSTR;
            const float f = mat[trow + k];
            for (int j = 0; j < N_; ++j) {
                if (j != k) mat[trow + j] = fmaf(-f, mat[krow + j], mat[trow + j]);
            }
            mat[trow + k] = -f * ip;
        } else {
            mat[krow + k] = ip;
        }
        __syncthreads();
    }

    // dB = dt * inv @ B   (row per thread; stride-257 rows are conflict-free)
    {
        float s = 0.0f;
        const int trow = tid * MSTR;
        for (int j = 0; j < N_; ++j) s = fmaf(mat[trow + j], bvec[j], s);
        xv[tid] = dt * s;
    }
    __syncthreads();

    // mat <- dA = 2*inv - I ; stream dA to global for the squaring chain
    float* __restrict__ dAh = dA_out + (size_t)h * N_ * N_;
    for (int idx = tid; idx < N_ * N_; idx += 256) {
        const int i = idx >> 8, j = idx & 255;
        const float v = 2.0f * mat[i * MSTR + j] - (i == j ? 1.0f : 0.0f);
        mat[i * MSTR + j] = v;
        dAh[idx] = v;
    }
    __syncthreads();

    // First Krylov block: X0[:,l] = dA^l dB, l = 0..15, plus k[h, 0..15].
    float* __restrict__ X0h  = X0_out + (size_t)h * N_ * 16;
    float* __restrict__ outh = out + (size_t)h * L_;
    for (int l = 0; l < 16; ++l) {
        X0h[tid * 16 + l] = xv[tid];
        red[tid] = cvec[tid] * xv[tid];
        float nx = 0.0f;
        const int trow = tid * MSTR;
        for (int j = 0; j < N_; ++j) nx = fmaf(mat[trow + j], xv[j], nx);
        __syncthreads();
        xv[tid] = nx;                               // all reads of xv are done
        for (int s2 = 128; s2 > 0; s2 >>= 1) {
            if (tid < s2) red[tid] += red[tid + s2];
            __syncthreads();
        }
        if (tid == 0) outh[l] = red[0];
        __syncthreads();
    }
}

// ---------------------------------------------------------------------------
// Kernel B: per-head D = S @ S (N x N) with V_WMMA_F32_16X16X4_F32.
// Grid (4,4,H): each WG computes a 64x64 tile; 8 waves x 2 16x16 subtiles.
// ---------------------------------------------------------------------------
__global__ __launch_bounds__(256) void matsq_kernel(
    const float* __restrict__ S, float* __restrict__ D)
{
    __shared__ float As[64 * 65];
    __shared__ float Bs[64 * 65];

    const int h = blockIdx.z, bx = blockIdx.x, by = blockIdx.y;
    const int tid  = threadIdx.x;
    const int wave = tid >> 5, lane = tid & 31;
    const int half = lane >> 4, lr = lane & 15;
    const int sub0 = wave * 2, sub1 = sub0 + 1;
    const int sm0 = (sub0 >> 2) * 16, sn0 = (sub0 & 3) * 16;
    const int sm1 = (sub1 >> 2) * 16, sn1 = (sub1 & 3) * 16;

    const float* __restrict__ Sh = S + (size_t)h * N_ * N_;
    v8f acc0 = {0.f,0.f,0.f,0.f,0.f,0.f,0.f,0.f};
    v8f acc1 = {0.f,0.f,0.f,0.f,0.f,0.f,0.f,0.f};

    for (int k0 = 0; k0 < N_; k0 += 64) {
        for (int idx = tid; idx < 4096; idx += 256) {
            const int r = idx >> 6, c = idx & 63;
            As[r * 65 + c] = Sh[(by * 64 + r) * N_ + k0 + c];
            Bs[r * 65 + c] = Sh[(k0 + r) * N_ + bx * 64 + c];
        }
        __syncthreads();
        for (int k2 = 0; k2 < 64; k2 += 4) {
            const int ka = k2 + 2 * half;           // K pair for this lane half
            v2f a0, b0, a1, b1;
            a0.x = As[(sm0 + lr) * 65 + ka];
            a0.y = As[(sm0 + lr) * 65 + ka + 1];
            b0.x = Bs[ka * 65 + sn0 + lr];
            b0.y = Bs[(ka + 1) * 65 + sn0 + lr];
            acc0 = __builtin_amdgcn_wmma_f32_16x16x4_f32(
                       false, a0, false, b0, (short)0, acc0, false, false);
            a1.x = As[(sm1 + lr) * 65 + ka];
            a1.y = As[(sm1 + lr) * 65 + ka + 1];
            b1.x = Bs[ka * 65 + sn1 + lr];
            b1.y = Bs[(ka + 1) * 65 + sn1 + lr];
            acc1 = __builtin_amdgcn_wmma_f32_16x16x4_f32(
                       false, a1, false, b1, (short)0, acc1, false, false);
        }
        __syncthreads();
    }

    float* __restrict__ Dh = D + (size_t)h * N_ * N_;
    #pragma unroll
    for (int v = 0; v < 8; ++v) {
        Dh[(by * 64 + sm0 + v + 8 * half) * N_ + bx * 64 + sn0 + lr] = acc0[v];
        Dh[(by * 64 + sm1 + v + 8 * half) * N_ + bx * 64 + sn1 + lr] = acc1[v];
    }
}

// ---------------------------------------------------------------------------
// Kernel C: per-head block recurrence. P = dA^16 resident in LDS (263 KB),
// 63 serial block steps X <- P @ X (256x256 @ 256x16, all WMMA f32),
// then k[h, 16*blk + i] = C . X[:, i].
// ---------------------------------------------------------------------------
__global__ __launch_bounds__(256) void krylov_kernel(
    const float* __restrict__ P, const float* __restrict__ X0,
    const float* __restrict__ Cv, float* __restrict__ out)
{
    __shared__ float Ps[N_ * MSTR];
    __shared__ float Xa[N_ * XSTR];
    __shared__ float Xb[N_ * XSTR];
    __shared__ float cvec[N_];
    __shared__ float red[N_];

    const int h = blockIdx.x, tid = threadIdx.x;
    const float* __restrict__ Ph  = P + (size_t)h * N_ * N_;
    const float* __restrict__ X0h = X0 + (size_t)h * N_ * 16;

    for (int idx = tid; idx < N_ * N_; idx += 256) {
        const int i = idx >> 8, j = idx & 255;
        Ps[i * MSTR + j] = Ph[idx];
    }
    for (int idx = tid; idx < N_ * 16; idx += 256) {
        const int r = idx >> 4, c = idx & 15;
        Xa[r * XSTR + c] = X0h[idx];
    }
    cvec[tid] = Cv[h * N_ + tid];
    __syncthreads();

    const int wave = tid >> 5, lane = tid & 31;
    const int half = lane >> 4, lr = lane & 15;
    const int sm0 = wave * 32, sm1 = wave * 32 + 16;   // 2 vertical subtiles/wave

    float* cur = Xa;
    float* nxt = Xb;
    float* __restrict__ outh = out + (size_t)h * L_;

    for (int blk = 1; blk < 64; ++blk) {
        v8f acc0 = {0.f,0.f,0.f,0.f,0.f,0.f,0.f,0.f};
        v8f acc1 = {0.f,0.f,0.f,0.f,0.f,0.f,0.f,0.f};
        for (int k = 0; k < N_; k += 4) {
            const int ka = k + 2 * half;
            v2f a0, a1, b0;
            b0.x = cur[ka * XSTR + lr];                // shared B operand
            b0.y = cur[(ka + 1) * XSTR + lr];
            a0.x = Ps[(sm0 + lr) * MSTR + ka];
            a0.y = Ps[(sm0 + lr) * MSTR + ka + 1];
            acc0 = __builtin_amdgcn_wmma_f32_16x16x4_f32(
                       false, a0, false, b0, (short)0, acc0, false, false);
            a1.x = Ps[(sm1 + lr) * MSTR + ka];
            a1.y = Ps[(sm1 + lr) * MSTR + ka + 1];
            acc1 = __builtin_amdgcn_wmma_f32_16x16x4_f32(
                       false, a1, false, b0, (short)0, acc1, false, false);
        }
        #pragma unroll
        for (int v = 0; v < 8; ++v) {
            nxt[(sm0 + v + 8 * half) * XSTR + lr] = acc0[v];
            nxt[(sm1 + v + 8 * half) * XSTR + lr] = acc1[v];
        }
        __syncthreads();

        // y[16*blk + i] = cvec . nxt[:, i] : 16 chunks x 16 columns
        {
            const int i = tid & 15, ch = tid >> 4;
            float p = 0.0f;
            for (int r = 0; r < 16; ++r) {
                const int row = ch * 16 + r;
                p = fmaf(cvec[row], nxt[row * XSTR + i], p);
            }
            red[tid] = p;
        }
        __syncthreads();
        if (tid < 16) {
            float y = 0.0f;
            for (int c = 0; c < 16; ++c) y += red[c * 16 + tid];
            outh[blk * 16 + tid] = y;
        }
        __syncthreads();

        float* t = cur; cur = nxt; nxt = t;
    }
}

// ---------------------------------------------------------------------------
extern "C" void kernel_launch(void* const* d_in, const int* in_sizes, int n_in,
                              void* d_out, int out_size, void* d_ws, size_t ws_size,
                              hipStream_t stream)
{
    const float* A     = (const float*)d_in[0];   // (H,N,N)
    const float* Bv    = (const float*)d_in[1];   // (H,N)
    const float* Cv    = (const float*)d_in[2];   // (1,H,N)
    const float* logdt = (const float*)d_in[3];   // (H,)
    float* out = (float*)d_out;                   // (1,H,L)

    const size_t matBytes = (size_t)H_ * N_ * N_ * sizeof(float);  // 64 MB
    float* buf0 = (float*)d_ws;                                    // dA / ping-pong
    float* buf1 = (float*)((char*)d_ws + matBytes);
    float* X0b  = (float*)((char*)d_ws + 2 * matBytes);            // (H,N,16)

    // Phase 1: invert + discretize + first 16 outputs; dA -> buf0, X0 -> X0b.
    disc_invert_kernel<<<H_, 256, 0, stream>>>(A, Bv, Cv, logdt, buf0, X0b, out);

    // Phase 2: P = dA^16 via 4 WMMA squarings (ping-pong), ends in buf0.
    dim3 g(4, 4, H_);
    matsq_kernel<<<g, 256, 0, stream>>>(buf0, buf1);   // dA^2
    matsq_kernel<<<g, 256, 0, stream>>>(buf1, buf0);   // dA^4
    matsq_kernel<<<g, 256, 0, stream>>>(buf0, buf1);   // dA^8
    matsq_kernel<<<g, 256, 0, stream>>>(buf1, buf0);   // dA^16

    // Phase 3: 63 WMMA block steps per head + output dots.
    krylov_kernel<<<H_, 256, 0, stream>>>(buf0, X0b, Cv, out);
}